// GCN_55095840473679
// MI455X (gfx1250) — compile-verified
//
#include <hip/hip_runtime.h>
#include <math.h>

typedef __attribute__((ext_vector_type(2))) float v2f;
typedef __attribute__((ext_vector_type(8))) float v8f;

#define TM 4   // 16-row tiles per wave in the WMMA GEMM (B-operand reuse factor)

// ---------------------------------------------------------------- utilities
__global__ void fill_kernel(float* __restrict__ p, float v, long long n) {
  long long i = (long long)blockIdx.x * blockDim.x + threadIdx.x;
  if (i < n) p[i] = v;
}

// deg[t] += 1 for every edge target (deg pre-initialized to 1.0 for self-loop)
__global__ void deg_kernel(const int* __restrict__ tgt, float* __restrict__ deg, int E) {
  int e = blockIdx.x * blockDim.x + threadIdx.x;
  if (e < E) atomicAdd(&deg[tgt[e]], 1.0f);
}

// dis = deg > 0 ? rsqrt(deg) : 0   (in place)
__global__ void rsqrt_kernel(float* __restrict__ d, int N) {
  int i = blockIdx.x * blockDim.x + threadIdx.x;
  if (i < N) {
    float v = d[i];
    d[i] = (v > 0.0f) ? rsqrtf(v) : 0.0f;
  }
}

// ------------------------------------------------------- layer-1 GEMM (WMMA)
// h[N,16] = x[N,D] @ W1[D,16]; one wave handles TM consecutive 16-row tiles so
// each B fragment (W1 K-slice) is loaded once and reused by TM WMMAs.
// A layout (ISA 7.12.2, 32-bit A 16x4): lane m = lane&15; VGPR pair holds
// K = kb, kb+1 with kb = 2*(lane>=16). C/D: VGPR i -> row i + 8*(lane>=16),
// col = lane&15.
__global__ __launch_bounds__(32) void gemm1_wmma_kernel(
    const float* __restrict__ x, const float* __restrict__ w,
    float* __restrict__ h, int D, int H) {
  const int lane = threadIdx.x & 31;
  const int row0 = blockIdx.x * (TM * 16);
  const int mn   = lane & 15;          // A row-in-tile / B,D col
  const int kb   = (lane >> 4) << 1;   // 0 or 2
  const float* __restrict__ wp = w + (size_t)kb * H + mn;  // per-lane W1 base
  v8f acc[TM];
#pragma unroll
  for (int j = 0; j < TM; ++j) acc[j] = (v8f){};

  for (int k0 = 0; k0 < D; k0 += 4) {
    v2f b;
    b.x = wp[(size_t)k0 * H];
    b.y = wp[(size_t)(k0 + 1) * H];
#pragma unroll
    for (int j = 0; j < TM; ++j) {
      const float* __restrict__ xr = x + (size_t)(row0 + j * 16 + mn) * D;
      v2f a;
      a.x = xr[k0 + kb];
      a.y = xr[k0 + kb + 1];
      acc[j] = __builtin_amdgcn_wmma_f32_16x16x4_f32(
          /*neg_a=*/false, a, /*neg_b=*/false, b,
          /*c_mod=*/(short)0, acc[j], /*reuse_a=*/false, /*reuse_b=*/false);
    }
  }

  const int mo = (lane >> 4) << 3;     // 0 or 8
#pragma unroll
  for (int j = 0; j < TM; ++j)
#pragma unroll
    for (int i = 0; i < 8; ++i)
      h[(size_t)(row0 + j * 16 + i + mo) * H + mn] = acc[j][i];
}

// scalar fallback / tail for rows not covered by the WMMA wave-groups
__global__ void gemm1_tail_kernel(const float* __restrict__ x, const float* __restrict__ w,
                                  float* __restrict__ h, int rowStart, int N, int D, int H) {
  long long idx = (long long)blockIdx.x * blockDim.x + threadIdx.x;
  long long total = (long long)(N - rowStart) * H;
  if (idx >= total) return;
  int r = rowStart + (int)(idx / H);
  int c = (int)(idx % H);
  float acc = 0.0f;
  for (int k = 0; k < D; ++k) acc += x[(size_t)r * D + k] * w[(size_t)k * H + c];
  h[(size_t)r * H + c] = acc;
}

// ------------------------------------------------ edge scatter, layer 1 (H%4==0)
__global__ void scatter1_kernel(const int* __restrict__ src, const int* __restrict__ tgt,
                                const float* __restrict__ dis, const float* __restrict__ h,
                                float* __restrict__ agg, long long total, int E, int H) {
  long long tid = (long long)blockIdx.x * blockDim.x + threadIdx.x;
  const int chunks = H >> 2;
  long long e = tid / chunks;
  if (e >= total) return;
  int f0 = (int)(tid % chunks) << 2;
  int s, t;
  if (e < E) { s = src[e]; t = tgt[e]; } else { s = t = (int)(e - E); }
  float nrm = dis[s] * dis[t];
  const float4 hv = *reinterpret_cast<const float4*>(h + (size_t)s * H + f0);
  float* dst = agg + (size_t)t * H + f0;
  atomicAdd(dst + 0, hv.x * nrm);
  atomicAdd(dst + 1, hv.y * nrm);
  atomicAdd(dst + 2, hv.z * nrm);
  atomicAdd(dst + 3, hv.w * nrm);
}

// fallback scatter (one feature per thread) for H not divisible by 4
__global__ void scatter1_scalar_kernel(const int* __restrict__ src, const int* __restrict__ tgt,
                                       const float* __restrict__ dis, const float* __restrict__ h,
                                       float* __restrict__ agg, long long total, int E, int H) {
  long long tid = (long long)blockIdx.x * blockDim.x + threadIdx.x;
  long long e = tid / H;
  if (e >= total) return;
  int f = (int)(tid % H);
  int s, t;
  if (e < E) { s = src[e]; t = tgt[e]; } else { s = t = (int)(e - E); }
  float nrm = dis[s] * dis[t];
  atomicAdd(&agg[(size_t)t * H + f], h[(size_t)s * H + f] * nrm);
}

// ----------------------------- fused bias + ReLU + layer-2 GEMM (per node)
__global__ void layer2_kernel(const float* __restrict__ agg1, const float* __restrict__ b1,
                              const float* __restrict__ W2, float* __restrict__ z,
                              int N, int H, int C) {
  int i = blockIdx.x * blockDim.x + threadIdx.x;
  if (i >= N) return;
  float acc[8];
  for (int c = 0; c < C; ++c) acc[c] = 0.0f;
  const float* __restrict__ row = agg1 + (size_t)i * H;
  for (int j = 0; j < H; ++j) {
    float v = row[j] + b1[j];
    v = v > 0.0f ? v : 0.0f;
    for (int c = 0; c < C; ++c) acc[c] += v * W2[(size_t)j * C + c];
  }
  for (int c = 0; c < C; ++c) z[(size_t)i * C + c] = acc[c];
}

// ------------------------------------------------ edge scatter, layer 2
__global__ void scatter2_kernel(const int* __restrict__ src, const int* __restrict__ tgt,
                                const float* __restrict__ dis, const float* __restrict__ z,
                                float* __restrict__ agg2, long long total, int E, int C) {
  long long e = (long long)blockIdx.x * blockDim.x + threadIdx.x;
  if (e >= total) return;
  int s, t;
  if (e < E) { s = src[e]; t = tgt[e]; } else { s = t = (int)(e - E); }
  float nrm = dis[s] * dis[t];
  for (int c = 0; c < C; ++c)
    atomicAdd(&agg2[(size_t)t * C + c], z[(size_t)s * C + c] * nrm);
}

// -------------------------------------- + b2, numerically stable log_softmax
__global__ void finalize_kernel(const float* __restrict__ agg2, const float* __restrict__ b2,
                                float* __restrict__ out, int N, int C) {
  int i = blockIdx.x * blockDim.x + threadIdx.x;
  if (i >= N) return;
  float l[8];
  float m = -INFINITY;
  for (int c = 0; c < C; ++c) {
    l[c] = agg2[(size_t)i * C + c] + b2[c];
    m = fmaxf(m, l[c]);
  }
  float ssum = 0.0f;
  for (int c = 0; c < C; ++c) ssum += expf(l[c] - m);
  float lse = m + logf(ssum);
  for (int c = 0; c < C; ++c) out[(size_t)i * C + c] = l[c] - lse;
}

// ---------------------------------------------------------------- launcher
extern "C" void kernel_launch(void* const* d_in, const int* in_sizes, int n_in,
                              void* d_out, int out_size, void* d_ws, size_t ws_size,
                              hipStream_t stream) {
  const float* x   = (const float*)d_in[0];
  const float* W1  = (const float*)d_in[1];
  const float* b1  = (const float*)d_in[2];
  const float* W2  = (const float*)d_in[3];
  const float* b2  = (const float*)d_in[4];
  const int*  edge = (const int*)d_in[5];

  const int H = in_sizes[2];
  const int C = in_sizes[4];
  const int D = in_sizes[1] / H;
  const int N = in_sizes[0] / D;
  const int E = in_sizes[5] / 2;
  const int* src = edge;       // edge_index[0]
  const int* tgt = edge + E;   // edge_index[1]

  float* ws   = (float*)d_ws;
  float* dis  = ws;                         // N      (deg -> rsqrt in place)
  float* h    = dis + (size_t)N;            // N*H
  float* agg1 = h + (size_t)N * H;          // N*H
  float* z    = agg1 + (size_t)N * H;       // N*C
  float* agg2 = z + (size_t)N * C;          // N*C
  float* out  = (float*)d_out;

  const long long total = (long long)E + (long long)N;   // edges + self-loops
  auto nblk = [](long long n, int t) { return (unsigned)((n + t - 1) / t); };

  // degree / normalization
  fill_kernel<<<nblk(N, 256), 256, 0, stream>>>(dis, 1.0f, N);   // self-loop count
  fill_kernel<<<nblk((long long)N * H, 256), 256, 0, stream>>>(agg1, 0.0f, (long long)N * H);
  fill_kernel<<<nblk((long long)N * C, 256), 256, 0, stream>>>(agg2, 0.0f, (long long)N * C);
  deg_kernel<<<nblk(E, 256), 256, 0, stream>>>(tgt, dis, E);
  rsqrt_kernel<<<nblk(N, 256), 256, 0, stream>>>(dis, N);

  // layer-1 GEMM: WMMA wave-groups of TM 16-row tiles when H==16 and D%4==0
  int nGroups = (H == 16 && (D & 3) == 0) ? (N / (16 * TM)) : 0;
  if (nGroups > 0)
    gemm1_wmma_kernel<<<nGroups, 32, 0, stream>>>(x, W1, h, D, H);
  int tailStart = nGroups * 16 * TM;
  if (tailStart < N) {
    long long tailElems = (long long)(N - tailStart) * H;
    gemm1_tail_kernel<<<nblk(tailElems, 256), 256, 0, stream>>>(x, W1, h, tailStart, N, D, H);
  }

  // layer-1 aggregation
  if ((H & 3) == 0)
    scatter1_kernel<<<nblk(total * (H >> 2), 256), 256, 0, stream>>>(
        src, tgt, dis, h, agg1, total, E, H);
  else
    scatter1_scalar_kernel<<<nblk(total * H, 256), 256, 0, stream>>>(
        src, tgt, dis, h, agg1, total, E, H);

  // bias + relu + layer-2 linear
  layer2_kernel<<<nblk(N, 256), 256, 0, stream>>>(agg1, b1, W2, z, N, H, C);

  // layer-2 aggregation
  scatter2_kernel<<<nblk(total, 256), 256, 0, stream>>>(src, tgt, dis, z, agg2, total, E, C);

  // bias + log_softmax
  finalize_kernel<<<nblk(N, 256), 256, 0, stream>>>(agg2, b2, out, N, C);
}